// DiscountedTypeLoss_87574383165820
// MI455X (gfx1250) — compile-verified
//
#include <hip/hip_runtime.h>
#include <hip/hip_bf16.h>
#include <math.h>

#define HIDDEN 1024
#define NUM_TAGS 16
#define TEMPERATURE 0.3f
#define EPS_CS 1e-8f

typedef float f4  __attribute__((ext_vector_type(4)));
typedef float v2f __attribute__((ext_vector_type(2)));
typedef float v8f __attribute__((ext_vector_type(8)));

// ---------------------------------------------------------------------------
// Kernel 1: stream features once (HBM-bound), accumulate per-class feature
// sums in registers. Label is wave-uniform per token -> scalar branch tree.
// Each thread owns 4 consecutive hidden channels (one b128 NT load / token).
// ---------------------------------------------------------------------------
__global__ __launch_bounds__(256) void k1_classsum(
    const float* __restrict__ features, const int* __restrict__ labels,
    float* __restrict__ part, float* __restrict__ pcnt, int N)
{
    __shared__ int s_cnt[NUM_TAGS];
    const int tid = threadIdx.x;
    if (tid < NUM_TAGS) s_cnt[tid] = 0;
    __syncthreads();

    f4 a0 = {}, a1 = {}, a2 = {}, a3 = {}, a4 = {}, a5 = {}, a6 = {}, a7 = {};
    f4 a8 = {}, a9 = {}, a10 = {}, a11 = {}, a12 = {}, a13 = {}, a14 = {}, a15 = {};

    const int hbase = tid * 4;
    for (int t = blockIdx.x; t < N; t += gridDim.x) {
        const f4* p = (const f4*)(features + (size_t)t * HIDDEN + hbase);
        f4 v = __builtin_nontemporal_load(p);
        int l = __builtin_amdgcn_readfirstlane(labels[t]);
        if (tid == 0) s_cnt[l] += 1;
        switch (l) {
            case 0:  a0  += v; break;
            case 1:  a1  += v; break;
            case 2:  a2  += v; break;
            case 3:  a3  += v; break;
            case 4:  a4  += v; break;
            case 5:  a5  += v; break;
            case 6:  a6  += v; break;
            case 7:  a7  += v; break;
            case 8:  a8  += v; break;
            case 9:  a9  += v; break;
            case 10: a10 += v; break;
            case 11: a11 += v; break;
            case 12: a12 += v; break;
            case 13: a13 += v; break;
            case 14: a14 += v; break;
            default: a15 += v; break;
        }
    }
    __syncthreads();

    float* outp = part + (size_t)blockIdx.x * NUM_TAGS * HIDDEN + hbase;
#define ST(i, acc) *(f4*)(outp + (size_t)(i) * HIDDEN) = acc;
    ST(0, a0)  ST(1, a1)  ST(2, a2)   ST(3, a3)   ST(4, a4)   ST(5, a5)
    ST(6, a6)  ST(7, a7)  ST(8, a8)   ST(9, a9)   ST(10, a10) ST(11, a11)
    ST(12, a12) ST(13, a13) ST(14, a14) ST(15, a15)
#undef ST
    if (tid < NUM_TAGS)
        pcnt[(size_t)blockIdx.x * NUM_TAGS + tid] = (float)s_cnt[tid];
}

// ---------------------------------------------------------------------------
// Kernel 2: deterministic reduction of per-block partials.
// 64 blocks x 256 threads = 16384 threads -> one (c,h) element each.
// ---------------------------------------------------------------------------
__global__ __launch_bounds__(256) void k2_reduce(
    const float* __restrict__ part, const float* __restrict__ pcnt,
    float* __restrict__ red, float* __restrict__ rcnt, int nblocks)
{
    const int i = blockIdx.x * 256 + threadIdx.x;   // [0, 16384)
    float s = 0.0f;
    for (int b = 0; b < nblocks; ++b)
        s += part[(size_t)b * (NUM_TAGS * HIDDEN) + i];
    red[i] = s;
    if (i < NUM_TAGS) {
        float s2 = 0.0f;
        for (int b = 0; b < nblocks; ++b)
            s2 += pcnt[(size_t)b * NUM_TAGS + i];
        rcnt[i] = s2;
    }
}

// ---------------------------------------------------------------------------
// Kernel 3: single wave32. WMMA f32 16x16x4 over K=1024:
//   D[c][t] = sum_h sums_feat[c][h] * W[t][h]
// A (16x4, MxK): lane m=lane&15, kb=(lane>>4)*2, v = {A[m][k+kb], A[m][k+kb+1]}
// B (4x16, KxN): lane n=lane&15, same kb,      v = {B[kb][n], B[kb+1][n]} = W[n][k+kb..]
// C/D (16x16 f32): vgpr j -> M = j + (lane<16 ? 0 : 8), N = lane&15.
// Then the 16x16 epilogue: one class per lane (0..15).
// ---------------------------------------------------------------------------
__global__ __launch_bounds__(32) void k3_finish(
    const float* __restrict__ red, const float* __restrict__ rcnt,
    const float* __restrict__ Wg, const float* __restrict__ bias,
    const float* __restrict__ proto, float* __restrict__ out)
{
    __shared__ float s_mat[NUM_TAGS * NUM_TAGS];
    __shared__ float s_proto[NUM_TAGS * NUM_TAGS];
    __shared__ float s_cntv[NUM_TAGS];
    __shared__ float s_loss[NUM_TAGS];

    const int lane = threadIdx.x;
    const int m  = lane & 15;
    const int kb = (lane >> 4) << 1;

    const float* arow = red + (size_t)m * HIDDEN + kb;
    const float* brow = Wg  + (size_t)m * HIDDEN + kb;

    v8f acc = {};
    for (int k = 0; k < HIDDEN; k += 4) {
        v2f av, bv;
        av.x = arow[k]; av.y = arow[k + 1];
        bv.x = brow[k]; bv.y = brow[k + 1];
        acc = __builtin_amdgcn_wmma_f32_16x16x4_f32(
                  false, av, false, bv, (short)0, acc, false, false);
    }

    const int mb = (lane < 16) ? 0 : 8;
#pragma unroll
    for (int j = 0; j < 8; ++j)
        s_mat[(mb + j) * NUM_TAGS + m] = acc[j];

    for (int i = lane; i < NUM_TAGS * NUM_TAGS; i += 32) s_proto[i] = proto[i];
    if (lane < NUM_TAGS) s_cntv[lane] = rcnt[lane];
    __syncthreads();

    if (lane < NUM_TAGS) {
        const int c = lane;
        const float cn = s_cntv[c];
        const float denom = fmaxf(cn, 1.0f);

        float mrow[16];
#pragma unroll
        for (int t = 0; t < 16; ++t)
            mrow[t] = (s_mat[c * 16 + t] + cn * bias[t]) / denom;

        float msq = 0.0f;
#pragma unroll
        for (int t = 0; t < 16; ++t) msq += mrow[t] * mrow[t];
        const float mn = fmaxf(sqrtf(msq), EPS_CS);

        float pn[16];
        float pnc = EPS_CS;
#pragma unroll
        for (int t = 0; t < 16; ++t) {
            float s = 0.0f;
#pragma unroll
            for (int k2 = 0; k2 < 16; ++k2) {
                const float pv = s_proto[t * 16 + k2];
                s += pv * pv;
            }
            pn[t] = fmaxf(sqrtf(s), EPS_CS);
            if (t == c) pnc = pn[t];
        }

        float logit[16], simr[16];
#pragma unroll
        for (int t = 0; t < 16; ++t) {
            float dot = 0.0f, sdot = 0.0f;
#pragma unroll
            for (int k2 = 0; k2 < 16; ++k2) {
                const float pv = s_proto[t * 16 + k2];
                dot  += mrow[k2] * pv;
                sdot += s_proto[c * 16 + k2] * pv;
            }
            const float cosv = dot / (mn * pn[t]);
            logit[t] = -(1.0f - cosv) / TEMPERATURE;
            simr[t]  = sdot / (pnc * pn[t]);
        }

        // rank[j] = #{j2 : sim[j2] > sim[j] or (== and j2 < j)}  (stable argsort)
        float diag = 0.0f;
#pragma unroll
        for (int t = 0; t < 16; ++t) {
            int r = 0;
#pragma unroll
            for (int j = 0; j < 16; ++j)
                r += (int)((simr[j] > simr[t]) || (simr[j] == simr[t] && j < t));
            logit[t] = logit[t] / log2f((float)r + 2.0f);
            if (t == c) diag = logit[t];
        }

        float mx = -3.4e38f;
#pragma unroll
        for (int t = 0; t < 16; ++t) mx = fmaxf(mx, logit[t]);
        float se = 0.0f;
#pragma unroll
        for (int t = 0; t < 16; ++t) se += expf(logit[t] - mx);
        const float lossc = -(diag - (logf(se) + mx));
        s_loss[c] = (cn > 0.0f) ? lossc : 0.0f;
    }
    __syncthreads();

    if (lane == 0) {
        float s = 0.0f;
#pragma unroll
        for (int t = 0; t < 16; ++t) s += s_loss[t];
        out[0] = s / (float)NUM_TAGS;
    }
}

// ---------------------------------------------------------------------------
extern "C" void kernel_launch(void* const* d_in, const int* in_sizes, int n_in,
                              void* d_out, int out_size, void* d_ws, size_t ws_size,
                              hipStream_t stream)
{
    const float* features = (const float*)d_in[0];
    const int*   labels   = (const int*)d_in[1];
    const float* Wg       = (const float*)d_in[2];
    const float* bias     = (const float*)d_in[3];
    const float* proto    = (const float*)d_in[4];
    const int N = in_sizes[1];   // B*S tokens

    // Workspace layout (floats): [nblocks*16*1024 partial sums]
    //                            [nblocks*16 partial counts]
    //                            [16*1024 reduced sums] [16 reduced counts]
    const size_t perBlk = (size_t)NUM_TAGS * HIDDEN + NUM_TAGS;
    const size_t fixed  = (size_t)NUM_TAGS * HIDDEN + NUM_TAGS;
    const size_t avail  = ws_size / sizeof(float);
    int nblocks = 512;
    while (nblocks > 1 && (size_t)nblocks * perBlk + fixed > avail) nblocks >>= 1;

    float* part = (float*)d_ws;
    float* pcnt = part + (size_t)nblocks * NUM_TAGS * HIDDEN;
    float* red  = pcnt + (size_t)nblocks * NUM_TAGS;
    float* rcnt = red  + (size_t)NUM_TAGS * HIDDEN;

    k1_classsum<<<nblocks, 256, 0, stream>>>(features, labels, part, pcnt, N);
    k2_reduce  <<<64,      256, 0, stream>>>(part, pcnt, red, rcnt, nblocks);
    k3_finish  <<<1,        32, 0, stream>>>(red, rcnt, Wg, bias, proto, (float*)d_out);
}